// SIGReg_32770600469032
// MI455X (gfx1250) — compile-verified
//
#include <hip/hip_runtime.h>
#include <hip/hip_bf16.h>
#include <math.h>

typedef __attribute__((ext_vector_type(2))) float v2f;
typedef __attribute__((ext_vector_type(8))) float v8f;
typedef __attribute__((ext_vector_type(4))) unsigned int v4u;
typedef __attribute__((ext_vector_type(8))) int v8i;
typedef __attribute__((ext_vector_type(4))) int v4i;

#define B_ROWS 8192
#define D_DIM  2048
#define K_DIM  512
#define T_PTS  17

// ---------------- workspace layout (floats) ----------------
#define PROJ_ELEMS (B_ROWS * K_DIM)          // 4,194,304
#define OFF_ZSUM  0                          // 2048
#define OFF_ZSS   (OFF_ZSUM + D_DIM)         // 2048
#define OFF_DSS   (OFF_ZSS + D_DIM)          // 512
#define OFF_DINV  (OFF_DSS + K_DIM)          // 512
#define OFF_PSUM  (OFF_DINV + K_DIM)         // 512
#define OFF_PSS   (OFF_PSUM + K_DIM)         // 512
#define OFF_PMEAN (OFF_PSS + K_DIM)          // 512
#define OFF_PINV  (OFF_PMEAN + K_DIM)        // 512
#define OFF_CF    (OFF_PINV + K_DIM)         // 17*2*512
#define OFF_VF    (OFF_CF + T_PTS * 2 * K_DIM)
#define TAIL_ELEMS (OFF_VF + 1)              // 24577

// ---------------- packed f32 VOP3P helpers (2 rows per lane) ----------------
__device__ inline v2f pk_mul(v2f a, v2f b) {
    v2f d;
    asm("v_pk_mul_f32 %0, %1, %2" : "=v"(d) : "v"(a), "v"(b));
    return d;
}
__device__ inline v2f pk_fma(v2f a, v2f b, v2f c) {       // a*b + c
    v2f d;
    asm("v_pk_fma_f32 %0, %1, %2, %3" : "=v"(d) : "v"(a), "v"(b), "v"(c));
    return d;
}
__device__ inline v2f pk_fms(v2f a, v2f b, v2f c) {       // a*b - c
    v2f d;
    asm("v_pk_fma_f32 %0, %1, %2, %3 neg_lo:[0,0,1] neg_hi:[0,0,1]"
        : "=v"(d) : "v"(a), "v"(b), "v"(c));
    return d;
}
__device__ inline v2f pk_add(v2f a, v2f b) {
    v2f d;
    asm("v_pk_add_f32 %0, %1, %2" : "=v"(d) : "v"(a), "v"(b));
    return d;
}

// ---------------- zero the accumulator tail ----------------
__global__ void zero_kernel(float* p, int n) {
    int i = blockIdx.x * blockDim.x + threadIdx.x;
    if (i < n) p[i] = 0.0f;
}

// ---------------- per-column sum / sumsq ----------------
__global__ __launch_bounds__(256) void colstats_kernel(const float* __restrict__ M,
                                                       int ncols, int rows_per_blk,
                                                       float* __restrict__ sum,
                                                       float* __restrict__ ss) {
    int col = blockIdx.x * 256 + threadIdx.x;
    int r0  = blockIdx.y * rows_per_blk;
    float s = 0.0f, q = 0.0f;
    const float* p = M + (size_t)r0 * ncols + col;
    for (int r = 0; r < rows_per_blk; ++r) {
        float v = *p;
        s += v;
        q += v * v;
        p += ncols;
    }
    if (sum) atomicAdd(&sum[col], s);
    atomicAdd(&ss[col], q);
}

// ---------------- finalize 1: var_floor + direction inv-norms ----------------
__global__ __launch_bounds__(256) void finalize1_kernel(const float* __restrict__ zsum,
                                                        const float* __restrict__ zss,
                                                        const float* __restrict__ dss,
                                                        float* __restrict__ dinv,
                                                        float* __restrict__ vf) {
    __shared__ float red[256];
    int tid = threadIdx.x;
    float local = 0.0f;
    for (int d = tid; d < D_DIM; d += 256) {
        float s = zsum[d], q = zss[d];
        float var = (q - s * s * (1.0f / (float)B_ROWS)) * (1.0f / (float)(B_ROWS - 1));
        float sd = sqrtf(fmaxf(var, 0.0f));
        float r = 1.0f - sd;               // VAR_GAMMA = 1.0
        local += (r > 0.0f) ? r : 0.0f;
    }
    red[tid] = local;
    __syncthreads();
    for (int s2 = 128; s2 > 0; s2 >>= 1) {
        if (tid < s2) red[tid] += red[tid + s2];
        __syncthreads();
    }
    if (tid == 0) vf[0] = red[0] * (1.0f / (float)D_DIM);
    for (int k = tid; k < K_DIM; k += 256) {
        float n = sqrtf(dss[k]);
        dinv[k] = 1.0f / fmaxf(n, 1e-12f);
    }
}

// ---------------- fp32 WMMA GEMM with TDM-staged A tile ----------------
// proj = Z (8192x2048) @ Dir (2048x512), column-scaled by dinv.
#define BM 128
#define BN 64
#define BK 16
#define AS 18     // A row stride in LDS: TDM pads 16 data DWORDs + 2 pad DWORDs
#define BPS 160   // B K-pair stride (floats): lanes16-31 land on banks 32..63

__global__ __launch_bounds__(256) void gemm_wmma_kernel(const float* __restrict__ Z,
                                                        const float* __restrict__ Dir,
                                                        const float* __restrict__ dinv,
                                                        float* __restrict__ proj) {
    __shared__ __align__(16) float Al[BM * AS];        // 2304 floats (TDM fills incl. pad)
    __shared__ __align__(16) float Bl[(BK / 2) * BPS]; // 1280 floats, K-pair interleaved

    int tid  = threadIdx.x;
    int lane = tid & 31;
    int w    = tid >> 5;          // 8 waves
    int wm   = w & 3;             // 4 waves over M
    int wn   = w >> 2;            // 2 waves over N
    int bm   = blockIdx.x * BM;
    int bn   = blockIdx.y * BN;
    int lhi  = lane >> 4;
    int llo  = lane & 15;

    // LDS byte offset of Al (flat-address low 32 bits are the LDS offset)
    unsigned lds_a = (unsigned)(uintptr_t)(void*)Al;

    v8f acc[2][2] = {};

    for (int k0 = 0; k0 < D_DIM; k0 += BK) {
        __syncthreads();   // previous tiles fully consumed before overwrite

        // --- A tile via Tensor Data Mover: 2D tile 16(k) x 128(m), pad 16+2 DWORDs ---
        if (w == 0) {
            unsigned long long ga = (unsigned long long)(uintptr_t)Z +
                                    ((unsigned long long)bm * D_DIM + (unsigned)k0) * 4ull;
            v4u g0;
            g0[0] = 1u;                                        // count=1 (valid D#)
            g0[1] = lds_a;                                     // lds_addr
            g0[2] = (unsigned)(ga & 0xffffffffull);            // global_addr[31:0]
            g0[3] = (unsigned)((ga >> 32) & 0x1ffffffull)      // global_addr[56:32]
                    | (2u << 30);                              // type = 2 (image)
            v8i g1;
            g1[0] = (2 << 16)      // data_size = 4B
                  | (1 << 20)      // pad_enable
                  | (3 << 22)      // pad_interval: 16 DWORDs
                  | (1 << 25);     // pad_amount: 2 DWORDs  -> row stride 18
            g1[1] = (int)((unsigned)D_DIM << 16);   // tensor_dim0 = 2048 (bits 63:48)
            g1[2] = (int)((unsigned)B_ROWS << 16);  // tensor_dim1 = 8192 (bits 95:80)
            g1[3] = (BK << 16);                     // tile_dim0 = 16    (bits 127:112)
            g1[4] = BM;                             // tile_dim1 = 128   (bits 143:128)
            g1[5] = D_DIM;                          // tensor_dim0_stride = 2048
            g1[6] = 0;
            g1[7] = 0;
            v4i g2 = {0, 0, 0, 0};
            v4i g3 = {0, 0, 0, 0};
            v8i g4 = {0, 0, 0, 0, 0, 0, 0, 0};      // unused trailing group (2D tensor)
            __builtin_amdgcn_tensor_load_to_lds(g0, g1, g2, g3, g4, 0);
        }

        // --- B tile 16x64 staged manually, K-pair interleaved: Bl[pair][col*2 + (k&1)] ---
        {
            int r = tid >> 4;                 // k-row 0..15
            int cbase = (tid & 15) * 4;       // col 0..60
            float4 v = *(const float4*)(Dir + (size_t)(k0 + r) * K_DIM + bn + cbase);
            float* bp = Bl + (r >> 1) * BPS + (r & 1);
            bp[(cbase + 0) * 2] = v.x;
            bp[(cbase + 1) * 2] = v.y;
            bp[(cbase + 2) * 2] = v.z;
            bp[(cbase + 3) * 2] = v.w;
        }

        if (w == 0) __builtin_amdgcn_s_wait_tensorcnt(0);   // A tile landed in LDS
        __syncthreads();                                     // publish A (via wave0) + B

        #pragma unroll
        for (int kk = 0; kk < BK; kk += 4) {
            int kb = kk + 2 * lhi;            // even K-pair base for this half-wave
            v2f a[2], b[2];
            #pragma unroll
            for (int tm = 0; tm < 2; ++tm) {
                int row = wm * 32 + tm * 16 + llo;
                a[tm] = *(const v2f*)(Al + row * AS + kb);             // one ds_load_b64
            }
            #pragma unroll
            for (int tn = 0; tn < 2; ++tn) {
                int col = wn * 32 + tn * 16 + llo;
                b[tn] = *(const v2f*)(Bl + (kb >> 1) * BPS + col * 2); // one ds_load_b64
            }
            #pragma unroll
            for (int tm = 0; tm < 2; ++tm)
                #pragma unroll
                for (int tn = 0; tn < 2; ++tn)
                    acc[tm][tn] = __builtin_amdgcn_wmma_f32_16x16x4_f32(
                        false, a[tm], false, b[tn], (short)0, acc[tm][tn], false, false);
        }
    }

    // Epilogue: fold 1/||d_k||, store C per ISA layout (M = v + 8*lhi, N = llo)
    #pragma unroll
    for (int tm = 0; tm < 2; ++tm) {
        #pragma unroll
        for (int tn = 0; tn < 2; ++tn) {
            int col = bn + wn * 32 + tn * 16 + llo;
            float sc = dinv[col];
            int rbase = bm + wm * 32 + tm * 16 + 8 * lhi;
            #pragma unroll
            for (int v = 0; v < 8; ++v)
                proj[(size_t)(rbase + v) * K_DIM + col] = acc[tm][tn][v] * sc;
        }
    }
}

// ---------------- finalize 2: proj column mean & 1/(std+1e-8) ----------------
__global__ __launch_bounds__(256) void finalize2_kernel(const float* __restrict__ psum,
                                                        const float* __restrict__ pss,
                                                        float* __restrict__ pmean,
                                                        float* __restrict__ pinv) {
    int k = blockIdx.x * 256 + threadIdx.x;
    if (k < K_DIM) {
        float s = psum[k], q = pss[k];
        float mean = s * (1.0f / (float)B_ROWS);
        float var = (q - s * s * (1.0f / (float)B_ROWS)) * (1.0f / (float)(B_ROWS - 1));
        float sd = sqrtf(fmaxf(var, 0.0f));
        pmean[k] = mean;
        pinv[k] = 1.0f / (sd + 1e-8f);
    }
}

// ---------------- CF pass: rotation recurrence, 2 rows/lane packed f32 ----------------
__global__ __launch_bounds__(256) void cf_kernel(const float* __restrict__ proj,
                                                 const float* __restrict__ pmean,
                                                 const float* __restrict__ pinv,
                                                 float* __restrict__ cf) {
    int k  = blockIdx.x * 256 + threadIdx.x;   // gridDim.x = 2
    int r0 = blockIdx.y * 128;                 // gridDim.y = 64
    float mean = pmean[k];
    float inv  = pinv[k];
    const float dt = 2.0f / (float)T_PTS;

    v2f aR[T_PTS], aI[T_PTS];
    #pragma unroll
    for (int t = 0; t < T_PTS; ++t) { aR[t] = (v2f)(0.0f); aI[t] = (v2f)(0.0f); }

    const float* p = proj + (size_t)r0 * K_DIM + k;
    for (int r = 0; r < 128; r += 2) {
        float x0 = (p[0] - mean) * inv;
        float x1 = (p[K_DIM] - mean) * inv;
        p += 2 * K_DIM;
        v2f c1, s1;
        c1.x = __cosf(dt * x0);  c1.y = __cosf(dt * x1);
        s1.x = __sinf(dt * x0);  s1.y = __sinf(dt * x1);
        v2f c = (v2f)(1.0f), s = (v2f)(0.0f);
        #pragma unroll
        for (int t = 0; t < T_PTS; ++t) {
            v2f cn = pk_fms(c, c1, pk_mul(s, s1));   // c*c1 - s*s1
            v2f sn = pk_fma(s, c1, pk_mul(c, s1));   // s*c1 + c*s1
            c = cn; s = sn;
            aR[t] = pk_add(aR[t], c);
            aI[t] = pk_add(aI[t], s);
        }
    }
    #pragma unroll
    for (int t = 0; t < T_PTS; ++t) {
        atomicAdd(&cf[(t * 2 + 0) * K_DIM + k], aR[t].x + aR[t].y);
        atomicAdd(&cf[(t * 2 + 1) * K_DIM + k], aI[t].x + aI[t].y);
    }
}

// ---------------- final scalar: quadrature + var_floor ----------------
__global__ __launch_bounds__(256) void final_kernel(const float* __restrict__ cf,
                                                    const float* __restrict__ vf,
                                                    float* __restrict__ out) {
    __shared__ float red[256];
    int tid = threadIdx.x;
    const float dt = 2.0f / (float)T_PTS;
    float local = 0.0f;
    for (int idx = tid; idx < T_PTS * K_DIM; idx += 256) {
        int t = idx / K_DIM;
        int k = idx - t * K_DIM;
        float er = cf[(t * 2 + 0) * K_DIM + k] * (1.0f / (float)B_ROWS);
        float ei = cf[(t * 2 + 1) * K_DIM + k] * (1.0f / (float)B_ROWS);
        float tv = dt * (float)(t + 1);
        float tcf = __expf(-0.5f * tv * tv);
        float integ = er * er + ei * ei - 2.0f * er * tcf + tcf * tcf;
        float w = (t == 0 || t == T_PTS - 1) ? (dt * 0.5f) : dt;
        local += w * integ * (1.0f / (float)K_DIM);
    }
    red[tid] = local;
    __syncthreads();
    for (int s2 = 128; s2 > 0; s2 >>= 1) {
        if (tid < s2) red[tid] += red[tid + s2];
        __syncthreads();
    }
    if (tid == 0) out[0] = red[0] + vf[0];
}

extern "C" void kernel_launch(void* const* d_in, const int* in_sizes, int n_in,
                              void* d_out, int out_size, void* d_ws, size_t ws_size,
                              hipStream_t stream) {
    const float* Z   = (const float*)d_in[0];   // (8192, 2048)
    const float* Dir = (const float*)d_in[1];   // (2048, 512)
    float* out = (float*)d_out;

    float* ws    = (float*)d_ws;
    float* proj  = ws;
    float* tail  = ws + PROJ_ELEMS;
    float* zsum  = tail + OFF_ZSUM;
    float* zss   = tail + OFF_ZSS;
    float* dss   = tail + OFF_DSS;
    float* dinv  = tail + OFF_DINV;
    float* psum  = tail + OFF_PSUM;
    float* pss   = tail + OFF_PSS;
    float* pmean = tail + OFF_PMEAN;
    float* pinv  = tail + OFF_PINV;
    float* cf    = tail + OFF_CF;
    float* vf    = tail + OFF_VF;

    zero_kernel<<<(TAIL_ELEMS + 255) / 256, 256, 0, stream>>>(tail, TAIL_ELEMS);
    colstats_kernel<<<dim3(D_DIM / 256, B_ROWS / 128), 256, 0, stream>>>(Z, D_DIM, 128, zsum, zss);
    colstats_kernel<<<dim3(K_DIM / 256, D_DIM / 128), 256, 0, stream>>>(Dir, K_DIM, 128, nullptr, dss);
    finalize1_kernel<<<1, 256, 0, stream>>>(zsum, zss, dss, dinv, vf);
    gemm_wmma_kernel<<<dim3(B_ROWS / BM, K_DIM / BN), 256, 0, stream>>>(Z, Dir, dinv, proj);
    colstats_kernel<<<dim3(K_DIM / 256, B_ROWS / 128), 256, 0, stream>>>(proj, K_DIM, 128, psum, pss);
    finalize2_kernel<<<dim3((K_DIM + 255) / 256), 256, 0, stream>>>(psum, pss, pmean, pinv);
    cf_kernel<<<dim3(K_DIM / 256, B_ROWS / 128), 256, 0, stream>>>(proj, pmean, pinv, cf);
    final_kernel<<<1, 256, 0, stream>>>(cf, vf, out);
}